// SpatialAdaptiveSynBatchNorm2d_noEdge_6485400617348
// MI455X (gfx1250) — compile-verified
//
#include <hip/hip_runtime.h>

typedef __attribute__((ext_vector_type(2))) float v2f;
typedef __attribute__((ext_vector_type(8))) float v8f;

#define B_   8
#define O_   8
#define C_   128
#define HW_  16384
#define NW_  512
#define NSAMP 131072.0f   // b*h*w per channel

// workspace layout (floats)
#define WS_PSUM   0                    // [C_][B_] partial sums
#define WS_PSQ    (C_*B_)              // [C_][B_] partial sumsq
#define WS_MEAN   (2*C_*B_)            // [C_]
#define WS_INVSTD (2*C_*B_ + C_)       // [C_]
#define WS_WT     (2*C_*B_ + 2*C_)     // [B_*O_][C_]
#define WS_BS     (WS_WT + B_*O_*C_)   // [B_*O_][C_]

// ---------------- Pass 1: per-(channel, batch) slab moments ----------------
__global__ void __launch_bounds__(256) k_stats(const float* __restrict__ x,
                                               float* __restrict__ ws) {
  __shared__ float ssum[256], ssq[256];
  const int c = blockIdx.x, b = blockIdx.y, t = threadIdx.x;
  const float* p = x + ((size_t)(b * C_ + c)) * HW_;
  float s = 0.f, q = 0.f;
  for (int i = t; i < HW_; i += 256) { float v = p[i]; s += v; q += v * v; }
  ssum[t] = s; ssq[t] = q; __syncthreads();
  for (int st = 128; st > 0; st >>= 1) {
    if (t < st) { ssum[t] += ssum[t + st]; ssq[t] += ssq[t + st]; }
    __syncthreads();
  }
  if (t == 0) {
    ws[WS_PSUM + c * B_ + b] = ssum[0];
    ws[WS_PSQ  + c * B_ + b] = ssq[0];
  }
}

// ---------------- Pass 2: finalize mean / invstd ----------------
__global__ void k_finalize(float* __restrict__ ws) {
  const int c = threadIdx.x;   // 128 threads
  float s = 0.f, q = 0.f;
  #pragma unroll
  for (int b = 0; b < B_; ++b) { s += ws[WS_PSUM + c * B_ + b]; q += ws[WS_PSQ + c * B_ + b]; }
  const float mean = s / NSAMP;
  const float var  = q / NSAMP - mean * mean;
  ws[WS_MEAN + c]   = mean;
  ws[WS_INVSTD + c] = rsqrtf(var + 1e-5f);
}

// ---------------- Pass 3: wt/bs = vector @ {w,b}_weight.T + bias (WMMA) ----
// grid: 32 blocks x 32 threads (1 wave). Tile (16m x 16n), K-loop over 512.
__global__ void __launch_bounds__(32) k_params(const float* __restrict__ vec,
                                               const float* __restrict__ ww,
                                               const float* __restrict__ wb,
                                               const float* __restrict__ bwt,
                                               const float* __restrict__ bb,
                                               float* __restrict__ ws) {
  const int lane = threadIdx.x;
  const int tm = blockIdx.x >> 3;       // 0..3  (M = 64)
  const int tn = blockIdx.x & 7;        // 0..7  (N = 128)
  const int lo = lane & 15;
  const int hi = (lane >> 4) & 1;
  const int koff = hi ? 2 : 0;
  const int mrow = tm * 16 + lo;        // A row (vector row)
  const int ncol = tn * 16 + lo;        // B col (channel)
  const float* arow = vec + (size_t)mrow * NW_;
  const float* wrow = ww  + (size_t)ncol * NW_;   // B[k][n] = ww[n*NW_+k]
  const float* brow = bwt + (size_t)ncol * NW_;
  v8f accW = {}; v8f accB = {};
  for (int k0 = 0; k0 < NW_; k0 += 4) {
    v2f a, fw, fb;
    a[0]  = arow[k0 + koff];     a[1]  = arow[k0 + koff + 1];
    fw[0] = wrow[k0 + koff];     fw[1] = wrow[k0 + koff + 1];
    fb[0] = brow[k0 + koff];     fb[1] = brow[k0 + koff + 1];
    accW = __builtin_amdgcn_wmma_f32_16x16x4_f32(false, a, false, fw, (short)0, accW, false, false);
    accB = __builtin_amdgcn_wmma_f32_16x16x4_f32(false, a, false, fb, (short)0, accB, false, false);
  }
  const float wbias = wb[ncol], bbias = bb[ncol];
  float* wt = ws + WS_WT; float* bs = ws + WS_BS;
  #pragma unroll
  for (int d = 0; d < 8; ++d) {
    const int row = tm * 16 + d + hi * 8;
    wt[row * C_ + ncol] = accW[d] + wbias;
    bs[row * C_ + ncol] = accB[d] + bbias;
  }
}

// ---------------- Pass 4: fused bbox-mix (WMMA) + batchnorm apply ----------
// block = (b, 256-pixel tile); 8 waves; 128 wave-tasks of (16c x 16pix).
__global__ void __launch_bounds__(256) k_main(const float* __restrict__ x,
                                              const float* __restrict__ bbox,
                                              const float* __restrict__ ws,
                                              float* __restrict__ out) {
  __shared__ float bbS[O_][256];        // bbox rows (K-major for B frags)
  __shared__ float rdenS[256];          // 1 / (sum_o bbox + eps)
  __shared__ float wtS[C_ * O_];        // [c][o] -> A matrix (c rows, K=o)
  __shared__ float bsS[C_ * O_];
  __shared__ float meanS[C_], invS[C_];
  const int t = threadIdx.x;
  const int b = blockIdx.y;
  const int pixBase = blockIdx.x * 256;

  #pragma unroll
  for (int o = 0; o < O_; ++o)
    bbS[o][t] = bbox[((size_t)(b * O_ + o)) * HW_ + pixBase + t];
  const float* wt = ws + WS_WT; const float* bs = ws + WS_BS;
  for (int i = t; i < C_ * O_; i += 256) {
    const int c = i >> 3, o = i & 7;
    wtS[i] = wt[(b * O_ + o) * C_ + c];
    bsS[i] = bs[(b * O_ + o) * C_ + c];
  }
  if (t < C_) { meanS[t] = ws[WS_MEAN + t]; invS[t] = ws[WS_INVSTD + t]; }
  {
    float s = 0.f;
    #pragma unroll
    for (int o = 0; o < O_; ++o) s += bbS[o][t];
    rdenS[t] = 1.0f / (s + 1e-6f);
  }
  __syncthreads();

  const int lane = t & 31;
  const int wave = t >> 5;              // 0..7
  const int lo = lane & 15;
  const int hi = (lane >> 4) & 1;
  const int koff = hi ? 2 : 0;

  for (int task = wave; task < 128; task += 8) {   // uniform per wave: EXEC all-1s
    const int ct  = task >> 4;          // channel tile 0..7
    const int sub = task & 15;          // pixel subtile 0..15
    const int crowA = ct * 16 + lo;     // A-matrix row (channel)
    v8f W = {}; v8f Bv = {};
    #pragma unroll
    for (int kk = 0; kk < 2; ++kk) {    // K=8 as two K=4 WMMA steps
      const int k = kk * 4 + koff;
      v2f aw, ab, bf;
      aw[0] = wtS[crowA * O_ + k]; aw[1] = wtS[crowA * O_ + k + 1];
      ab[0] = bsS[crowA * O_ + k]; ab[1] = bsS[crowA * O_ + k + 1];
      bf[0] = bbS[k][sub * 16 + lo];
      bf[1] = bbS[k + 1][sub * 16 + lo];
      W  = __builtin_amdgcn_wmma_f32_16x16x4_f32(false, aw, false, bf, (short)0, W,  false, false);
      Bv = __builtin_amdgcn_wmma_f32_16x16x4_f32(false, ab, false, bf, (short)0, Bv, false, false);
    }
    const int pcol = sub * 16 + lo;
    const float rden = rdenS[pcol];
    const int pix = pixBase + pcol;
    #pragma unroll
    for (int d = 0; d < 8; ++d) {
      const int crow = ct * 16 + d + hi * 8;       // D layout: M = d (+8 upper half)
      const size_t idx = ((size_t)(b * C_ + crow)) * HW_ + pix;
      const float xn  = (x[idx] - meanS[crow]) * invS[crow];
      const float wgt = W[d] * rden + 1.0f;
      const float bia = Bv[d] * rden;
      out[idx] = wgt * xn + bia;
    }
  }
}

extern "C" void kernel_launch(void* const* d_in, const int* in_sizes, int n_in,
                              void* d_out, int out_size, void* d_ws, size_t ws_size,
                              hipStream_t stream) {
  (void)in_sizes; (void)n_in; (void)out_size; (void)ws_size;
  const float* x    = (const float*)d_in[0];
  const float* vec  = (const float*)d_in[1];
  const float* bbox = (const float*)d_in[2];
  const float* ww   = (const float*)d_in[3];
  const float* wb   = (const float*)d_in[4];
  const float* bwt  = (const float*)d_in[5];
  const float* bb   = (const float*)d_in[6];
  float* out = (float*)d_out;
  float* ws  = (float*)d_ws;

  k_stats   <<<dim3(C_, B_), 256, 0, stream>>>(x, ws);
  k_finalize<<<1, C_, 0, stream>>>(ws);
  k_params  <<<32, 32, 0, stream>>>(vec, ww, wb, bwt, bb, ws);
  k_main    <<<dim3(HW_ / 256, B_), 256, 0, stream>>>(x, bbox, ws, out);
}